// VectorQuantizer_41455024341497
// MI455X (gfx1250) — compile-verified
//
#include <hip/hip_runtime.h>
#include <hip/hip_bf16.h>

// VQ-VAE vector quantizer for MI455X (gfx1250, wave32).
//   latents  : [65536, 64] f32, codebook : [1024, 64] f32
//   out      : quantized [65536,64] f32 ++ vq_loss scalar f32
//
// dist(n,k) = |x|^2 + |e|^2 - 2 x.e ; argmin drops |x|^2.
// GEMM x.e^T on V_WMMA_F32_16X16X4_F32 (fp32, matches reference argmin).
// M = codes, N = latents -> per-lane accumulators hold 8 code candidates for
// ONE latent => lane-local argmin; halves merged with one shfl_xor(16).
// Codebook chunks stream L2 -> LDS via GLOBAL_LOAD_ASYNC_TO_LDS_B128
// (ASYNCcnt), double-buffered so the copy overlaps the WMMA chain.

#define NLAT   65536
#define DDIM   64
#define KCODE  1024
#define KC     64             // codes per LDS chunk (double-buffered)
#define NCHUNK (KCODE / KC)   // 16
#define CB_STRIDE 68          // padded row stride (floats): conflict-free A frags

typedef float v2f __attribute__((ext_vector_type(2)));
typedef float v8f __attribute__((ext_vector_type(8)));

// ---------------------------------------------------------------- norms ----
__global__ __launch_bounds__(256) void vq_norms_kernel(
    const float* __restrict__ cb, float* __restrict__ norms) {
  int c = blockIdx.x * 256 + threadIdx.x;
  if (c < KCODE) {
    const float* row = cb + (size_t)c * DDIM;
    float s = 0.f;
    #pragma unroll
    for (int d = 0; d < DDIM; d += 4) {
      float4 v = *(const float4*)(row + d);
      s = fmaf(v.x, v.x, s);
      s = fmaf(v.y, v.y, s);
      s = fmaf(v.z, v.z, s);
      s = fmaf(v.w, v.w, s);
    }
    norms[c] = s;
  }
}

// Issue async L2->LDS copies for one 64-code chunk (4 x b128 per thread).
// ASYNCcnt-tracked; no VGPR data path, no ds-store traffic.
__device__ __forceinline__ void stage_chunk_async(
    const float* __restrict__ cb, int c0, unsigned ldsBase, int tid) {
  #pragma unroll
  for (int i = 0; i < 4; ++i) {
    int f4  = i * 256 + tid;                 // float4 index in chunk (0..1023)
    int row = f4 >> 4;                       // 16 float4 per 64-float row
    int c4  = f4 & 15;
    const float* g = cb + (size_t)(c0 + row) * DDIM + c4 * 4;
    unsigned l = ldsBase + (unsigned)((row * CB_STRIDE + c4 * 4) * 4);
    asm volatile("global_load_async_to_lds_b128 %0, %1, off"
                 :: "v"(l), "v"(g) : "memory");
  }
}

// ----------------------------------------------------------------- main ----
__global__ __launch_bounds__(256) void vq_main_kernel(
    const float* __restrict__ lat, const float* __restrict__ cb,
    const float* __restrict__ norms, float* __restrict__ qout,
    float* __restrict__ partials) {
  __shared__ float sCB[2][KC * CB_STRIDE];   // double-buffered chunk (~34 KB)
  __shared__ float sNorm[KCODE];             // all code norms (4 KB)
  __shared__ float sRed[8];

  const int tid    = threadIdx.x;
  const int wave   = tid >> 5;
  const int lane   = tid & 31;
  const int laneLo = lane & 15;
  const int hi     = lane >> 4;              // 0: K={0,1}, 1: K={2,3} per 4-step
  const int dOff   = 2 * hi;
  const int n      = blockIdx.x * 128 + wave * 16 + laneLo;

  const unsigned ldsBuf0 = (unsigned)(size_t)&sCB[0][0];  // LDS offset = addr[31:0]
  const unsigned ldsBuf1 = (unsigned)(size_t)&sCB[1][0];

  // Prologue: async-stage chunk 0, load all norms once, preload B fragments.
  stage_chunk_async(cb, 0, ldsBuf0, tid);
  *(float4*)&sNorm[tid * 4] = *(const float4*)(norms + tid * 4);

  // B fragments (latents), register-resident for ALL code tiles.
  // Assumed B(4x16): VGPRr = K=r (lanes 0-15) / K=r+2 (lanes 16-31), N=lane&15.
  v2f bfrag[16];
  const float* xrow = lat + (size_t)n * DDIM;
  #pragma unroll
  for (int s = 0; s < 16; ++s)
    bfrag[s] = *(const v2f*)(xrow + 4 * s + dOff);

  float bestScore = 3.4e38f;
  int   bestIdx   = 0;

  asm volatile("s_wait_asynccnt 0x0" ::: "memory");
  __syncthreads();

  for (int chunk = 0; chunk < NCHUNK; ++chunk) {
    const int cur = chunk & 1;
    const float* sBuf = &sCB[cur][0];
    // Prefetch next chunk into the other buffer while we compute.
    if (chunk + 1 < NCHUNK)
      stage_chunk_async(cb, (chunk + 1) * KC, cur ? ldsBuf0 : ldsBuf1, tid);

    const int c0 = chunk * KC;
    #pragma unroll
    for (int ct = 0; ct < KC / 16; ++ct) {   // 4 code tiles per chunk
      v8f acc = {0.f, 0.f, 0.f, 0.f, 0.f, 0.f, 0.f, 0.f};
      // A(16x4) per ISA: M=lane&15; VGPR0/1 = K0/K1 (lanes 0-15), K2/K3 (16-31)
      const float* aBase = sBuf + (ct * 16 + laneLo) * CB_STRIDE + dOff;
      #pragma unroll
      for (int s = 0; s < 16; ++s) {         // D=64 in 16 K-steps of 4
        v2f a = *(const v2f*)(aBase + 4 * s);
        acc = __builtin_amdgcn_wmma_f32_16x16x4_f32(
            false, a, false, bfrag[s], (short)0, acc, false, false);
      }
      const int cbase = c0 + ct * 16 + 8 * hi;
      #pragma unroll
      for (int r = 0; r < 8; ++r) {
        float score = fmaf(-2.0f, acc[r], sNorm[cbase - (8 * hi) + (8 * hi) + r]);
        if (score < bestScore) { bestScore = score; bestIdx = cbase + r; }
      }
    }

    // Own async issues done -> barrier makes the staged buffer visible to all.
    asm volatile("s_wait_asynccnt 0x0" ::: "memory");
    __syncthreads();
  }

  // Merge half-lane candidates (lower code index wins ties, matching argmin).
  {
    float oS = __shfl_xor(bestScore, 16, 32);
    int   oI = __shfl_xor(bestIdx, 16, 32);
    if (oS < bestScore || (oS == bestScore && oI < bestIdx)) {
      bestScore = oS; bestIdx = oI;
    }
  }

  // Emit quantized = codebook[best]; loss partial from register-resident x.
  float lsum = 0.f;
  const float* erow = cb + (size_t)bestIdx * DDIM;
  float* qrow = qout + (size_t)n * DDIM;
  #pragma unroll
  for (int s = 0; s < 16; ++s) {
    v2f e = *(const v2f*)(erow + 4 * s + dOff);
    *(v2f*)(qrow + 4 * s + dOff) = e;
    float d0 = e.x - bfrag[s].x;
    float d1 = e.y - bfrag[s].y;
    lsum = fmaf(d0, d0, lsum);
    lsum = fmaf(d1, d1, lsum);
  }

  // Deterministic reduction: shuffle tree -> LDS -> per-WG partial.
  #pragma unroll
  for (int off = 16; off >= 1; off >>= 1)
    lsum += __shfl_xor(lsum, off, 32);
  if (lane == 0) sRed[wave] = lsum;
  __syncthreads();
  if (tid == 0) {
    float t = 0.f;
    #pragma unroll
    for (int w = 0; w < 8; ++w) t += sRed[w];
    partials[blockIdx.x] = t;
  }
}

// ------------------------------------------------------------- finalize ----
__global__ __launch_bounds__(32) void vq_finalize_kernel(
    const float* __restrict__ partials, float* __restrict__ outScalar) {
  int lane = threadIdx.x;
  float s = 0.f;
  #pragma unroll
  for (int j = 0; j < 512 / 32; ++j) s += partials[j * 32 + lane];
  #pragma unroll
  for (int off = 16; off >= 1; off >>= 1) s += __shfl_xor(s, off, 32);
  if (lane == 0)
    *outScalar = s * (1.25f / ((float)NLAT * (float)DDIM));   // (1+BETA)*mean
}

// --------------------------------------------------------------- launch ----
extern "C" void kernel_launch(void* const* d_in, const int* in_sizes, int n_in,
                              void* d_out, int out_size, void* d_ws, size_t ws_size,
                              hipStream_t stream) {
  const float* lat = (const float*)d_in[0];
  const float* cb  = (const float*)d_in[1];
  float* qout      = (float*)d_out;
  float* scalar    = qout + (size_t)NLAT * DDIM;      // tuple: [quantized..., loss]
  float* norms     = (float*)d_ws;                    // [1024]
  float* partials  = norms + KCODE;                   // [512]

  vq_norms_kernel<<<(KCODE + 255) / 256, 256, 0, stream>>>(cb, norms);
  vq_main_kernel<<<NLAT / 128, 256, 0, stream>>>(lat, cb, norms, qout, partials);
  vq_finalize_kernel<<<1, 32, 0, stream>>>(partials, scalar);
}